// ConvertParamMomentsTEtoParamsTE_60438779789426
// MI455X (gfx1250) — compile-verified
//
#include <hip/hip_runtime.h>

// ConvertParamMomentsTEtoParamsTE — MI455X (gfx1250, wave32, WMMA)
//
// B=4096, NH=64, NV=128. Memory-bound (~420 MB @ 23.3 TB/s => ~18us floor).
// Single fused streaming pass over varvhTE/varvh; h-reductions for bTE done
// with V_WMMA_F32_16X16X4_F32 so the B-operand-layout loads triple-serve the
// matvec, the elementwise wtTE, and the trace reduction. Non-temporal
// loads/stores for the streamed tensors (read/write-once, >> L2).

typedef float v2f __attribute__((ext_vector_type(2)));
typedef float v8f __attribute__((ext_vector_type(8)));

#define NH_ 64
#define NV_ 128

__global__ __launch_bounds__(256)
void cpm_te_kernel(const float* __restrict__ muvTE,       // [B,128]
                   const float* __restrict__ varvhTE,     // [B,64,128]
                   const float* __restrict__ varh_diagTE, // [B,64]
                   const float* __restrict__ varh_diag,   // [B,64]
                   const float* __restrict__ muh,         // [B,64]
                   const float* __restrict__ varvh,       // [B,64,128]
                   const float* __restrict__ muhTE,       // [B,64]
                   const float* __restrict__ varvbarTE,   // [B]
                   const int*   __restrict__ nvp,         // [1]
                   float* __restrict__ out_bTE,           // [B,128]
                   float* __restrict__ out_wtTE,          // [B,64,128]
                   float* __restrict__ out_sig2TE,        // [B]
                   float* __restrict__ out_muhTE,         // [B,64]
                   float* __restrict__ out_dTE)           // [B,64]
{
    const int b    = blockIdx.x;
    const int tid  = threadIdx.x;
    const int lane = tid & 31;
    const int wave = tid >> 5;   // 8 waves; wave w owns v-tile [16w, 16w+16)

    __shared__ float sA_TE[NH_]; // -vinv*muh                      (A coefs, varvhTE term)
    __shared__ float sA_V [NH_]; // vinv*(dTE*vinv*muh - muhTE)    (A coefs, varvh term)
    __shared__ float sW0 [NH_];  // -(vinv^2)*dTE                  (wtTE coef of varvh)
    __shared__ float sW1 [NH_];  // vinv                           (wtTE coef of varvhTE)
    __shared__ float sTr[8];

    if (tid < NH_) {
        const int   h   = tid;
        const float d   = varh_diagTE[b * NH_ + h];
        const float vi  = 1.0f / varh_diag[b * NH_ + h];
        const float m   = muh [b * NH_ + h];
        const float mTE = muhTE[b * NH_ + h];
        sA_TE[h] = -(vi * m);
        sA_V [h] = vi * (d * vi * m - mTE);
        sW0  [h] = -(vi * vi * d);
        sW1  [h] = vi;
        // passthrough outputs
        out_muhTE[b * NH_ + h] = mTE;
        out_dTE  [b * NH_ + h] = d;
    }
    __syncthreads();

    // WMMA operand striping (f32, K=4): VGPR slot j, half-wave sel -> K = j + 2*sel
    const int    sel   = lane >> 4;
    const int    vn    = lane & 15;
    const int    v0    = wave * 16;
    const size_t mbase = (size_t)b * NH_ * NV_;

    v8f   acc = {0.f, 0.f, 0.f, 0.f, 0.f, 0.f, 0.f, 0.f};
    float trp = 0.f;

    #pragma unroll 4
    for (int h0 = 0; h0 < NH_; h0 += 4) {
        const int    kA = h0 + 2 * sel;  // K index held in VGPR slot 0 by this lane
        const int    kB = kA + 1;        // K index held in VGPR slot 1
        const size_t iA = mbase + (size_t)kA * NV_ + v0 + vn;
        const size_t iB = mbase + (size_t)kB * NV_ + v0 + vn;

        // streamed, read-once -> non-temporal
        const float te0 = __builtin_nontemporal_load(varvhTE + iA);
        const float te1 = __builtin_nontemporal_load(varvhTE + iB);
        const float vv0 = __builtin_nontemporal_load(varvh   + iA);
        const float vv1 = __builtin_nontemporal_load(varvh   + iB);

        v2f aTE; aTE.x = sA_TE[kA]; aTE.y = sA_TE[kB];  // broadcast rows
        v2f aV;  aV.x  = sA_V [kA]; aV.y  = sA_V [kB];
        v2f bTEm; bTEm.x = te0; bTEm.y = te1;           // B operand = tensor tile
        v2f bVm;  bVm.x  = vv0; bVm.y  = vv1;

        // acc[m,n] += sum_k coef[h0+k] * M[h0+k, v0+n]   (all rows m identical)
        acc = __builtin_amdgcn_wmma_f32_16x16x4_f32(false, aTE, false, bTEm,
                                                    (short)0, acc, false, false);
        acc = __builtin_amdgcn_wmma_f32_16x16x4_f32(false, aV,  false, bVm,
                                                    (short)0, acc, false, false);

        // wtTE elementwise on the same registers
        const float w0A = sW0[kA], w1A = sW1[kA];
        const float w0B = sW0[kB], w1B = sW1[kB];
        const float wt0 = w0A * vv0 + w1A * te0;
        const float wt1 = w0B * vv1 + w1B * te1;
        __builtin_nontemporal_store(wt0, out_wtTE + iA);
        __builtin_nontemporal_store(wt1, out_wtTE + iB);

        // trace element = varvh * (wtTE + w1*varvhTE)  (= w0*vv^2 + 2*w1*te*vv)
        trp += vv0 * (wt0 + w1A * te0) + vv1 * (wt1 + w1B * te1);
    }

    // D row M=0 lives in accumulator VGPR 0, lanes 0..15 (N = lane)
    if (lane < 16) {
        const size_t ib = (size_t)b * NV_ + v0 + lane;
        out_bTE[ib] = muvTE[ib] + acc[0];
    }

    // trace: wave reduce, then cross-wave via LDS
    #pragma unroll
    for (int off = 16; off > 0; off >>= 1)
        trp += __shfl_xor(trp, off, 32);
    if (lane == 0) sTr[wave] = trp;
    __syncthreads();
    if (tid == 0) {
        float tr = 0.f;
        #pragma unroll
        for (int w = 0; w < 8; ++w) tr += sTr[w];
        out_sig2TE[b] = (varvbarTE[b] - tr) / (float)nvp[0];
    }
}

extern "C" void kernel_launch(void* const* d_in, const int* in_sizes, int n_in,
                              void* d_out, int out_size, void* d_ws, size_t ws_size,
                              hipStream_t stream) {
    const float* muvTE       = (const float*)d_in[0];
    const float* varvhTE     = (const float*)d_in[1];
    const float* varh_diagTE = (const float*)d_in[2];
    const float* varh_diag   = (const float*)d_in[3];
    const float* muh         = (const float*)d_in[4];
    const float* varvh       = (const float*)d_in[5];
    const float* muhTE       = (const float*)d_in[6];
    const float* varvbarTE   = (const float*)d_in[7];
    const int*   nvp         = (const int*)d_in[8];

    const int B = in_sizes[7];  // varvbarTE is [B]

    float* out        = (float*)d_out;
    float* out_bTE    = out;                                   // [B,128]
    float* out_wtTE   = out_bTE  + (size_t)B * NV_;            // [B,64,128]
    float* out_sig2TE = out_wtTE + (size_t)B * NH_ * NV_;      // [B]
    float* out_muhTE  = out_sig2TE + (size_t)B;                // [B,64]
    float* out_dTE    = out_muhTE  + (size_t)B * NH_;          // [B,64]

    cpm_te_kernel<<<B, 256, 0, stream>>>(muvTE, varvhTE, varh_diagTE, varh_diag,
                                         muh, varvh, muhTE, varvbarTE, nvp,
                                         out_bTE, out_wtTE, out_sig2TE,
                                         out_muhTE, out_dTE);
}